// ConstrastiveLoss2_70549132804621
// MI455X (gfx1250) — compile-verified
//
#include <hip/hip_runtime.h>

// ContrastiveLoss (CLIP-style) for B=4096, D=512 on gfx1250.
// loss = (1/2B) * sum_i [ LSE_row(i) + LSE_col(i) - 2*logits[i,i] ],
// logits = tau * F @ R^T.  Col-LSE(F@R^T) == Row-LSE(R@F^T), so one
// fused fp32-WMMA GEMM + online-LSE kernel is launched twice.

#define BN 4096
#define DK 512
#define SA_STRIDE 260   // padded per-lane stride (floats) to spread LDS banks

typedef __attribute__((ext_vector_type(2))) float v2f;
typedef __attribute__((ext_vector_type(4))) float v4f;
typedef __attribute__((ext_vector_type(8))) float v8f;

// ---------------------------------------------------------------------------
// Fused GEMM (fp32 WMMA 16x16x4) + per-row online log-sum-exp.
// Workgroup: 128 threads (4 waves), owns 16 rows of A (staged+packed in LDS).
// Wave w sweeps a 1024-column range as 32 iterations of a 16x32 output tile:
// each ds_load_b128 A-fragment feeds 4 WMMAs (two column sub-tiles).
// ---------------------------------------------------------------------------
__launch_bounds__(128)
__global__ void rowlse_kernel(const float* __restrict__ A,
                              const float* __restrict__ Bm,
                              const float* __restrict__ log_tau,
                              float* __restrict__ lse_out) {
    __shared__ __align__(16) float sA[32 * SA_STRIDE];  // packed A fragments (~33 KB)
    __shared__ float redM[4][8][32];                    // per-lane LSE partials
    __shared__ float redS[4][8][32];

    const int tid  = threadIdx.x;
    const int lane = tid & 31;
    const int wave = tid >> 5;
    const int m0   = blockIdx.x * 16;
    const int h    = lane >> 4;   // K-half select within a K-quad
    const int l16  = lane & 15;   // row (for A) / column (for B/C) index

    const float tau   = fminf(expf(log_tau[0]), 100.0f);
    const float tauL2 = tau * 1.4426950408889634f;      // fold log2(e): work in base-2

    // ---- stage & pack A tile (16 rows x 512 K) into WMMA A-fragment order ----
    // A[m0+r][k] -> sA[dl*SA_STRIDE + (k>>2)*2 + (k&1)], dl = r + 16*((k>>1)&1)
    for (int it = 0; it < 64; ++it) {
        int i = tid + 128 * it;           // k-major -> coalesced global reads
        int r = i >> 9;
        int k = i & 511;
        float val = A[(size_t)(m0 + r) * DK + k];
        int dl = r + (((k >> 1) & 1) << 4);
        sA[dl * SA_STRIDE + ((k >> 2) << 1) + (k & 1)] = val;
    }
    __syncthreads();

    // online LSE state: sm[v]/ss[v] cover row (v + 8*h), this lane's column slice
    float sm[8], ss[8];
    #pragma unroll
    for (int v = 0; v < 8; ++v) { sm[v] = -INFINITY; ss[v] = 0.0f; }

    const float* sAp = sA + lane * SA_STRIDE;
    for (int t = 0; t < 32; ++t) {
        const int n0 = wave * 1024 + t * 32;
        const float* bp0 = Bm + (size_t)(n0 + l16) * DK + (h << 1);
        const float* bp1 = bp0 + (size_t)16 * DK;
        if (t + 1 < 32)  // hint the next B tile toward L2/WGP$
            __builtin_prefetch(Bm + (size_t)(n0 + 32 + l16) * DK + (h << 1), 0, 1);

        v8f c0 = {};
        v8f c1 = {};
        #pragma unroll 4
        for (int kk = 0; kk < 128; kk += 2) {
            v4f a4  = *(const v4f*)(sAp + kk * 2);      // ds_load_b128: 2 k-steps of A
            v2f b00 = *(const v2f*)(bp0 + kk * 4);      // B frags, sub-tile 0
            v2f b01 = *(const v2f*)(bp0 + kk * 4 + 4);
            v2f b10 = *(const v2f*)(bp1 + kk * 4);      // B frags, sub-tile 1
            v2f b11 = *(const v2f*)(bp1 + kk * 4 + 4);
            v2f a0 = { a4.x, a4.y };
            v2f a1 = { a4.z, a4.w };
            c0 = __builtin_amdgcn_wmma_f32_16x16x4_f32(false, a0, false, b00,
                                                       (short)0, c0, false, false);
            c1 = __builtin_amdgcn_wmma_f32_16x16x4_f32(false, a0, false, b10,
                                                       (short)0, c1, false, false);
            c0 = __builtin_amdgcn_wmma_f32_16x16x4_f32(false, a1, false, b01,
                                                       (short)0, c0, false, false);
            c1 = __builtin_amdgcn_wmma_f32_16x16x4_f32(false, a1, false, b11,
                                                       (short)0, c1, false, false);
        }

        // two new columns per (row,lane) state this iteration: base-2 online LSE
        #pragma unroll
        for (int v = 0; v < 8; ++v) {
            float y0 = c0[v] * tauL2;
            float nm = fmaxf(sm[v], y0);
            ss[v] = ss[v] * exp2f(sm[v] - nm) + exp2f(y0 - nm);
            sm[v] = nm;
            float y1 = c1[v] * tauL2;
            float nm2 = fmaxf(sm[v], y1);
            ss[v] = ss[v] * exp2f(sm[v] - nm2) + exp2f(y1 - nm2);
            sm[v] = nm2;
        }
    }

    // ---- combine the 4 waves x 16 lanes partials per row ----
    #pragma unroll
    for (int v = 0; v < 8; ++v) {
        redM[wave][v][lane] = sm[v];
        redS[wave][v][lane] = ss[v];
    }
    __syncthreads();

    if (tid < 16) {
        const int r = tid;
        const int vv = r & 7;
        const int lbase = (r >> 3) << 4;
        float m = -INFINITY, s = 0.0f;
        for (int w = 0; w < 4; ++w)
            for (int l = 0; l < 16; ++l) {
                float m2 = redM[w][vv][lbase + l];
                float s2 = redS[w][vv][lbase + l];
                float nm = fmaxf(m, m2);
                s = s * exp2f(m - nm) + s2 * exp2f(m2 - nm);
                m = nm;
            }
        // natural-log LSE = ln2*m + ln(s)
        lse_out[m0 + r] = 0.6931471805599453f * m + logf(s);
    }
}

// ---------------------------------------------------------------------------
// diag[i] = tau * dot(F[i], R[i]); one wave32 per row, shfl-xor reduction.
// ---------------------------------------------------------------------------
__launch_bounds__(256)
__global__ void diag_kernel(const float* __restrict__ A,
                            const float* __restrict__ Bm,
                            const float* __restrict__ log_tau,
                            float* __restrict__ diag) {
    const int lane = threadIdx.x & 31;
    const int wave = threadIdx.x >> 5;
    const int row  = blockIdx.x * 8 + wave;
    const float tau = fminf(expf(log_tau[0]), 100.0f);
    const float* a = A  + (size_t)row * DK + lane * 16;
    const float* b = Bm + (size_t)row * DK + lane * 16;
    float acc = 0.0f;
    #pragma unroll
    for (int i = 0; i < 4; ++i) {
        v4f x = *(const v4f*)(a + i * 4);
        v4f y = *(const v4f*)(b + i * 4);
        acc += x.x * y.x + x.y * y.y + x.z * y.z + x.w * y.w;
    }
    #pragma unroll
    for (int off = 16; off; off >>= 1)
        acc += __shfl_xor(acc, off, 32);
    if (lane == 0) diag[row] = tau * acc;
}

// ---------------------------------------------------------------------------
// loss = sum_i (lse_r + lse_c - 2*diag) / (2B)
// ---------------------------------------------------------------------------
__launch_bounds__(256)
__global__ void finalize_kernel(const float* __restrict__ lse_r,
                                const float* __restrict__ lse_c,
                                const float* __restrict__ diag,
                                float* __restrict__ out) {
    __shared__ float red[256];
    float acc = 0.0f;
    for (int i = threadIdx.x; i < BN; i += 256)
        acc += lse_r[i] + lse_c[i] - 2.0f * diag[i];
    red[threadIdx.x] = acc;
    __syncthreads();
    for (int s = 128; s; s >>= 1) {
        if (threadIdx.x < s) red[threadIdx.x] += red[threadIdx.x + s];
        __syncthreads();
    }
    if (threadIdx.x == 0) out[0] = red[0] / (2.0f * (float)BN);
}

extern "C" void kernel_launch(void* const* d_in, const int* in_sizes, int n_in,
                              void* d_out, int out_size, void* d_ws, size_t ws_size,
                              hipStream_t stream) {
    const float* ftir    = (const float*)d_in[0];
    const float* raman   = (const float*)d_in[1];
    // d_in[2] = labels (unused by the reference math)
    const float* log_tau = (const float*)d_in[3];

    float* ws    = (float*)d_ws;
    float* lse_r = ws;            // 4096
    float* lse_c = ws + BN;       // 4096
    float* dg    = ws + 2 * BN;   // 4096

    rowlse_kernel<<<BN / 16, 128, 0, stream>>>(ftir, raman, log_tau, lse_r);  // row LSE
    rowlse_kernel<<<BN / 16, 128, 0, stream>>>(raman, ftir, log_tau, lse_c);  // col LSE
    diag_kernel<<<BN / 8, 256, 0, stream>>>(ftir, raman, log_tau, dg);
    finalize_kernel<<<1, 256, 0, stream>>>(lse_r, lse_c, dg, (float*)d_out);
}